// big_net_16853451669674
// MI455X (gfx1250) — compile-verified
//
#include <hip/hip_runtime.h>
#include <hip/hip_bf16.h>

// Problem constants (from reference): B=8, C=4, H=W=64, N=H*W=4096, 5x5 conv pad 2.
#define BATCH 8
#define CH    4
#define HW    64
#define NPIX  4096            // 64*64
#define IMG   (CH * NPIX)     // elements per batch of a [C,H,W] tensor

typedef __attribute__((ext_vector_type(2))) float v2f;
typedef __attribute__((ext_vector_type(8))) float v8f;

// ---------------------------------------------------------------------------
// Conv 5x5, pad 2, CIN input channels -> 4 output channels, optional sigmoid.
// One thread per output element; total = 8*4*64*64 = 131072 = 512 blocks x 256.
// This stage is ~100 MFLOP total: negligible vs the 512MB output stream.
// ---------------------------------------------------------------------------
template<int CIN, bool SIG>
__global__ __launch_bounds__(256) void conv5_kernel(const float* __restrict__ in,
                                                    const float* __restrict__ w,
                                                    const float* __restrict__ bias,
                                                    float* __restrict__ out) {
    int idx = blockIdx.x * 256 + threadIdx.x;        // exact grid, no tail
    int col =  idx        & 63;
    int row = (idx >> 6)  & 63;
    int co  = (idx >> 12) & 3;
    int b   =  idx >> 14;

    float acc = bias[co];
    const float* wc = w  + co * CIN * 25;
    const float* ib = in + b  * CIN * NPIX;
    for (int ci = 0; ci < CIN; ++ci) {
        const float* ic = ib + ci * NPIX;
        const float* wk = wc + ci * 25;
#pragma unroll
        for (int ky = 0; ky < 5; ++ky) {
            int iy = row + ky - 2;
            if ((unsigned)iy >= 64u) continue;
#pragma unroll
            for (int kx = 0; kx < 5; ++kx) {
                int ix = col + kx - 2;
                if ((unsigned)ix >= 64u) continue;
                acc = fmaf(wk[ky * 5 + kx], ic[iy * 64 + ix], acc);
            }
        }
    }
    if (SIG) acc = 1.0f / (1.0f + __expf(-acc));
    out[idx] = acc;
}

// ---------------------------------------------------------------------------
// Per-pixel channel stats: mean over C=4, unbiased std (ddof=1), centered Xc,
// and R = 1/(sqrt(C)*std) = 0.5/std so that R_i*R_j = 1/(C*S_i*S_j).
// One thread per (b, pixel); total = 8*4096 = 32768 = 128 blocks x 256.
// ---------------------------------------------------------------------------
__global__ __launch_bounds__(256) void stats_kernel(const float* __restrict__ y,
                                                    float* __restrict__ Xc,
                                                    float* __restrict__ R) {
    int idx = blockIdx.x * 256 + threadIdx.x;        // exact grid
    int n = idx & (NPIX - 1);
    int b = idx >> 12;
    const float* yb = y + b * IMG + n;
    float v0 = yb[0 * NPIX], v1 = yb[1 * NPIX], v2 = yb[2 * NPIX], v3 = yb[3 * NPIX];
    float mean = 0.25f * (v0 + v1 + v2 + v3);
    float d0 = v0 - mean, d1 = v1 - mean, d2 = v2 - mean, d3 = v3 - mean;
    float var = (d0 * d0 + d1 * d1 + d2 * d2 + d3 * d3) * (1.0f / 3.0f);  // ddof=1
    float s = __builtin_sqrtf(var);
    float* xb = Xc + b * IMG + n;
    xb[0 * NPIX] = d0;
    xb[1 * NPIX] = d1;
    xb[2 * NPIX] = d2;
    xb[3 * NPIX] = d3;
    R[idx] = 0.5f / s;                                // 1/(sqrt(4)*s)
}

// ---------------------------------------------------------------------------
// cov[b][i][j] = dot(Xc[:,i], Xc[:,j]) * R[i] * R[j]   via V_WMMA_F32_16X16X4_F32.
//
// One wave computes 4 consecutive 16x16 tiles along j (A tile reused).
// f32 16x4 A layout (ISA 7.12.2): lane%16 = M, lane-group selects K {0,1} vs {2,3}
// across the two VGPRs; the 4x16 B layout is the symmetric transpose, so both
// tiles load with identical per-lane addressing (column base differs).
// C/D layout: VGPR v -> row i0 + v + 8*(lane>=16), col j0 + lane%16.
// Output is 512MB write-once streaming -> non-temporal stores.
// Waves: 8 batches * 256 i-tiles * 64 j-groups = 131072 -> 16384 blocks x 8 waves.
// ---------------------------------------------------------------------------
__global__ __launch_bounds__(256) void cov_wmma_kernel(const float* __restrict__ Xc,
                                                       const float* __restrict__ R,
                                                       float* __restrict__ out) {
    const int wave = blockIdx.x * 8 + (threadIdx.x >> 5);
    const int lane = threadIdx.x & 31;

    const int b   =  wave >> 14;                 // 16384 waves per batch
    const int ti  = (wave >> 6) & 255;           // 256 i-tiles
    const int tjg =  wave       & 63;            // 64 groups of 4 j-tiles

    const int i0 = ti * 16;
    const int m  = lane & 15;
    const int kb = (lane < 16) ? 0 : 2;          // K base for this lane group

    const float* Xb = Xc + b * IMG;
    const float* Rb = R  + b * NPIX;
    float* outb = out + (size_t)b * NPIX * NPIX;

    // A tile: A[m][k] = Xc[k][i0+m]
    v2f a;
    a.x = Xb[(kb + 0) * NPIX + i0 + m];
    a.y = Xb[(kb + 1) * NPIX + i0 + m];

    const int rowbase = i0 + ((lane < 16) ? 0 : 8);
    float ri[8];
#pragma unroll
    for (int v = 0; v < 8; ++v) ri[v] = Rb[rowbase + v];

#pragma unroll
    for (int t = 0; t < 4; ++t) {
        const int j0 = (tjg * 4 + t) * 16;

        // B tile: B[k][n] = Xc[k][j0+n]  (same per-lane addressing as A)
        v2f bm;
        bm.x = Xb[(kb + 0) * NPIX + j0 + m];
        bm.y = Xb[(kb + 1) * NPIX + j0 + m];

        v8f c = {};
        c = __builtin_amdgcn_wmma_f32_16x16x4_f32(
                /*neg_a=*/false, a, /*neg_b=*/false, bm,
                /*c_mod=*/(short)0, c, /*reuse_a=*/false, /*reuse_b=*/false);

        const float rj = Rb[j0 + m];
#pragma unroll
        for (int v = 0; v < 8; ++v) {
            float val = c[v] * ri[v] * rj;
            __builtin_nontemporal_store(val,
                &outb[(size_t)(rowbase + v) * NPIX + j0 + m]);
        }
    }
}

// ---------------------------------------------------------------------------
// Launch: conv0(sig) -> conv1(sig) -> conv2(sig) -> conv3 -> stats -> cov.
// Workspace layout (floats): bufA[131072] | bufB[131072] | R[32768]  (~1.2 MB).
// ---------------------------------------------------------------------------
extern "C" void kernel_launch(void* const* d_in, const int* in_sizes, int n_in,
                              void* d_out, int out_size, void* d_ws, size_t ws_size,
                              hipStream_t stream) {
    const float* x  = (const float*)d_in[0];
    const float* w0 = (const float*)d_in[1];
    const float* b0 = (const float*)d_in[2];
    const float* w1 = (const float*)d_in[3];
    const float* b1 = (const float*)d_in[4];
    const float* w2 = (const float*)d_in[5];
    const float* b2 = (const float*)d_in[6];
    const float* w3 = (const float*)d_in[7];
    const float* b3 = (const float*)d_in[8];

    float* bufA = (float*)d_ws;                       // [8,4,4096]
    float* bufB = bufA + BATCH * IMG;                 // [8,4,4096]
    float* Rbuf = bufB + BATCH * IMG;                 // [8,4096]
    float* out  = (float*)d_out;                      // [8,4096,4096]

    const int convBlocks = (BATCH * IMG) / 256;       // 512

    conv5_kernel<1, true ><<<convBlocks, 256, 0, stream>>>(x,    w0, b0, bufA);
    conv5_kernel<CH, true ><<<convBlocks, 256, 0, stream>>>(bufA, w1, b1, bufB);
    conv5_kernel<CH, true ><<<convBlocks, 256, 0, stream>>>(bufB, w2, b2, bufA);
    conv5_kernel<CH, false><<<convBlocks, 256, 0, stream>>>(bufA, w3, b3, bufB);

    stats_kernel<<<(BATCH * NPIX) / 256, 256, 0, stream>>>(bufB, bufA, Rbuf);

    // 8 * 256 * 64 waves, 8 waves per 256-thread block -> 16384 blocks
    cov_wmma_kernel<<<16384, 256, 0, stream>>>(bufA, Rbuf, out);
}